// GRN_39573828665348
// MI455X (gfx1250) — compile-verified
//
#include <hip/hip_runtime.h>
#include <math.h>

// ---------------------------------------------------------------------------
// Problem constants (from reference): B=16, N=512, K=16, D=256, V=64, L=2
// ---------------------------------------------------------------------------
#define M_ROWS   8192
#define DIM      256
#define ZCOLS    1024
#define KNB      16
#define WT_MAT   262144      // 1024 cols * 256 k per transposed panel
#define AGG_ELEM 2097152     // 8192 * 256
#define TILE_M   64          // GEMM block tile rows (TDM tile height)

typedef __attribute__((ext_vector_type(2))) float v2f;
typedef __attribute__((ext_vector_type(4))) float v4f;
typedef __attribute__((ext_vector_type(8))) float v8f;
typedef __attribute__((ext_vector_type(4))) unsigned int u32x4;
typedef __attribute__((ext_vector_type(4))) int i32x4;
typedef __attribute__((ext_vector_type(8))) int i32x8;

#if __has_builtin(__builtin_amdgcn_tensor_load_to_lds)
#define HAVE_TDM 1
#else
#define HAVE_TDM 0
#endif

static __device__ __forceinline__ v8f wmma_f32(v2f a, v2f b, v8f c) {
  return __builtin_amdgcn_wmma_f32_16x16x4_f32(
      false, a, false, b, (short)0, c, false, false);
}

#if HAVE_TDM
// Issue one TDM DMA: 64 rows x 256 f32 (row stride 256 f32) -> contiguous LDS.
// D# per CDNA5 ISA ch.8: group0 = count/lds_addr/global_addr/type,
// group1 = data_size(4B), tensor_dim0=256, tensor_dim1=8192, tile 256x64,
// tensor_dim0_stride=256. Groups 2/3 unused (2D tensor).
static __device__ __forceinline__ void tdm_load_tile(unsigned lds_off,
                                                     const float* gptr) {
  unsigned long long ga = (unsigned long long)gptr;
  u32x4 g0 = { 1u,                                   // count=1 valid descriptor
               lds_off,                              // lds_addr (bytes)
               (unsigned)ga,                         // global_addr[31:0]
               ((unsigned)(ga >> 32) & 0x01FFFFFFu) | (2u << 30) };  // type=2
  i32x8 g1 = { (int)(2u << 16),        // data_size = 4 bytes
               (int)(256u << 16),      // tensor_dim0 = 256 (lo16 -> [31:16])
               (int)(8192u << 16),     // td0 hi=0 | tensor_dim1(8192) lo16
               (int)(256u << 16),      // td1 hi=0 | tile_dim0 = 256
               (int)TILE_M,            // tile_dim1 = 64 rows (tile_dim2 = 0)
               256,                    // tensor_dim0_stride lo32 = 256
               0, 0 };
  i32x4 zz = { 0, 0, 0, 0 };
#if defined(__clang_major__) && (__clang_major__ >= 23)
  i32x8 z8 = { 0, 0, 0, 0, 0, 0, 0, 0 };
  __builtin_amdgcn_tensor_load_to_lds(g0, g1, zz, zz, z8, 0);
#else
  __builtin_amdgcn_tensor_load_to_lds(g0, g1, zz, zz, 0);
#endif
}
#endif

// Stage a 64x256 f32 tile into LDS: TDM (async, wave 0) or cooperative copy.
static __device__ __forceinline__ void stage_issue(float* buf, unsigned lds_off,
                                                   const float* src,
                                                   int tid, int wave) {
#if HAVE_TDM
  if (wave == 0) tdm_load_tile(lds_off, src);
#else
  for (int t = tid; t < TILE_M * DIM / 4; t += 256)
    ((v4f*)buf)[t] = ((const v4f*)src)[t];
#endif
}

template <int N>
static __device__ __forceinline__ void stage_wait(int wave) {
#if HAVE_TDM
  if (wave == 0) __builtin_amdgcn_s_wait_tensorcnt(N);
#endif
}

// ---------------------------------------------------------------------------
// prep_wt: 6 transposed (col-major, K-contiguous) 1024x256 weight panels:
//   0: Wi_top^T  1: (Wi_bot+Ui_bot)^T  2: Wo_top^T  3: (Wo_bot+Uo_bot)^T
//   4: Ui_top^T  5: Uo_top^T
// ---------------------------------------------------------------------------
__global__ void prep_wt(const float* __restrict__ Wi, const float* __restrict__ Wo,
                        const float* __restrict__ Ui, const float* __restrict__ Uo,
                        float* __restrict__ WT) {
  int id = blockIdx.x * blockDim.x + threadIdx.x;
  if (id >= 6 * WT_MAT) return;
  int mat = id >> 18;
  int rem = id & (WT_MAT - 1);
  int col = rem >> 8;
  int k   = rem & 255;
  float v;
  switch (mat) {
    case 0:  v = Wi[k * ZCOLS + col]; break;
    case 1:  v = Wi[(k + 256) * ZCOLS + col] + Ui[(k + 256) * ZCOLS + col]; break;
    case 2:  v = Wo[k * ZCOLS + col]; break;
    case 3:  v = Wo[(k + 256) * ZCOLS + col] + Uo[(k + 256) * ZCOLS + col]; break;
    case 4:  v = Ui[k * ZCOLS + col]; break;
    default: v = Uo[k * ZCOLS + col]; break;
  }
  WT[id] = v;
}

__global__ void prep_bias(const float* __restrict__ bi, const float* __restrict__ bo,
                          const float* __restrict__ bui, const float* __restrict__ buo,
                          float* __restrict__ bs) {
  int i = blockIdx.x * blockDim.x + threadIdx.x;
  if (i < ZCOLS) bs[i] = bi[i] + bo[i] + bui[i] + buo[i];
}

// ---------------------------------------------------------------------------
// gather_agg: one wave per (b,n) row; 32 lanes x 8 floats = 256 = D.
// ---------------------------------------------------------------------------
__global__ void gather_agg(const float* __restrict__ h,
                           const int* __restrict__ iidx, const int* __restrict__ iedge,
                           const float* __restrict__ imask,
                           const int* __restrict__ oidx, const int* __restrict__ oedge,
                           const float* __restrict__ omask,
                           const float* __restrict__ emb,
                           float* __restrict__ Ain, float* __restrict__ Aout,
                           float* __restrict__ Ein, float* __restrict__ Eout,
                           int withE) {
  int wave = threadIdx.x >> 5;
  int lane = threadIdx.x & 31;
  int row  = blockIdx.x * 8 + wave;
  int baseb = (row >> 9) << 9;
  int off = lane * 8;

  const int*   ii = iidx  + row * KNB;
  const int*   ie = iedge + row * KNB;
  const float* im = imask + row * KNB;
  const int*   oi = oidx  + row * KNB;
  const int*   oe = oedge + row * KNB;
  const float* om = omask + row * KNB;

  v4f ai0 = {}, ai1 = {}, ao0 = {}, ao1 = {};
  v4f ei0 = {}, ei1 = {}, eo0 = {}, eo1 = {};

  for (int k = 0; k < KNB; ++k) {
    {
      int idx = ii[k]; float m = im[k];
      const v4f* hp = (const v4f*)(h + (baseb + idx) * DIM + off);
      ai0 += m * hp[0]; ai1 += m * hp[1];
      if (withE) {
        const v4f* ep = (const v4f*)(emb + ie[k] * DIM + off);
        ei0 += m * ep[0]; ei1 += m * ep[1];
      }
    }
    {
      int idx = oi[k]; float m = om[k];
      const v4f* hp = (const v4f*)(h + (baseb + idx) * DIM + off);
      ao0 += m * hp[0]; ao1 += m * hp[1];
      if (withE) {
        const v4f* ep = (const v4f*)(emb + oe[k] * DIM + off);
        eo0 += m * ep[0]; eo1 += m * ep[1];
      }
    }
  }
  v4f* pai = (v4f*)(Ain  + row * DIM + off); pai[0] = ai0; pai[1] = ai1;
  v4f* pao = (v4f*)(Aout + row * DIM + off); pao[0] = ao0; pao[1] = ao1;
  if (withE) {
    v4f* pei = (v4f*)(Ein  + row * DIM + off); pei[0] = ei0; pei[1] = ei1;
    v4f* peo = (v4f*)(Eout + row * DIM + off); peo[0] = eo0; peo[1] = eo1;
  }
}

// ---------------------------------------------------------------------------
// gemm_zconst: z_const = Ain@WT0 + Ein@WT1 + Aout@WT2 + Eout@WT3 + bias.
// Block: 64(M) x 64(N). 8 waves = (mw:2) x (nw:4); each wave 2 M-subtiles.
// A-tiles (64x256 = 64KB) staged in LDS by the Tensor Data Mover,
// double-buffered across the 4 passes (T0/T1 upfront, T2/T3 after WAR sync).
// ---------------------------------------------------------------------------
__global__ void gemm_zconst(const float* __restrict__ Ain, const float* __restrict__ Ein,
                            const float* __restrict__ Aout, const float* __restrict__ Eout,
                            const float* __restrict__ WT, const float* __restrict__ bias,
                            float* __restrict__ zc) {
  __shared__ float As[2][TILE_M * DIM];  // 2 x 64KB

  int tid = threadIdx.x;
  int wave = tid >> 5, lane = tid & 31;
  int mw = wave >> 2, nw = wave & 3;
  int hi = lane >> 4, l15 = lane & 15;
  int m0 = blockIdx.x * TILE_M, n0 = blockIdx.y * 64;
  int col = n0 + nw * 16 + l15;

  const float* Amats[4] = { Ain, Ein, Aout, Eout };
  const float* Bp[4] = { WT + col * DIM + 2 * hi,
                         WT + 1 * WT_MAT + col * DIM + 2 * hi,
                         WT + 2 * WT_MAT + col * DIM + 2 * hi,
                         WT + 3 * WT_MAT + col * DIM + 2 * hi };

  unsigned lds_off[2] = { (unsigned)(unsigned long long)&As[0][0],
                          (unsigned)(unsigned long long)&As[1][0] };

  stage_issue(As[0], lds_off[0], Amats[0] + (size_t)m0 * DIM, tid, wave);
  stage_issue(As[1], lds_off[1], Amats[1] + (size_t)m0 * DIM, tid, wave);

  int rA0 = (mw * 32 + l15) * DIM + 2 * hi;   // tile-local row, subtile 0
  int rA1 = rA0 + 16 * DIM;                   // subtile 1 (+16 rows)

  v8f acc0 = {}, acc1 = {};
  for (int p = 0; p < 4; ++p) {
    if (p == 3) stage_wait<0>(wave);
    else        stage_wait<1>(wave);
    __syncthreads();
    const float* a = &As[p & 1][0];
    const float* b = Bp[p];
    #pragma unroll 8
    for (int ks = 0; ks < DIM; ks += 4) {
      v2f bv = *(const v2f*)(b + ks);
      v2f a0 = *(const v2f*)(a + rA0 + ks);
      v2f a1 = *(const v2f*)(a + rA1 + ks);
      acc0 = wmma_f32(a0, bv, acc0);
      acc1 = wmma_f32(a1, bv, acc1);
    }
    __syncthreads();   // WAR: everyone done with buffer (p&1)
    if (p < 2)
      stage_issue(As[p & 1], lds_off[p & 1],
                  Amats[p + 2] + (size_t)m0 * DIM, tid, wave);
  }

  float bsum = bias[col];
  #pragma unroll
  for (int i = 0; i < 8; ++i) {
    int r0 = m0 + mw * 32 + i + 8 * hi;
    zc[r0 * ZCOLS + col] = acc0[i] + bsum;
    zc[(r0 + 16) * ZCOLS + col] = acc1[i] + bsum;
  }
}

// ---------------------------------------------------------------------------
// gemm_lstm: delta = Ain@Ui_top + Aout@Uo_top; z = z_const + delta; fused LSTM.
// Block: 64(M) x 64 cols per gate; wave holds 2 M-subtiles x 4 gates = 8 accs,
// so one LDS A-fragment pair feeds 8 WMMAs. A-tiles staged by TDM.
// ---------------------------------------------------------------------------
__global__ void gemm_lstm(const float* __restrict__ Ain, const float* __restrict__ Aout,
                          const float* __restrict__ WT, const float* __restrict__ zc,
                          const float* __restrict__ maskv,
                          const float* __restrict__ c_in, float* __restrict__ c_out,
                          float* __restrict__ h_out, float* __restrict__ h_out2,
                          int first) {
  __shared__ float As[2][TILE_M * DIM];

  int tid = threadIdx.x;
  int wave = tid >> 5, lane = tid & 31;
  int mw = wave >> 2, nw = wave & 3;
  int hi = lane >> 4, l15 = lane & 15;
  int m0 = blockIdx.x * TILE_M, n0 = blockIdx.y * 64;
  int colL = n0 + nw * 16 + l15;

  unsigned lds_off[2] = { (unsigned)(unsigned long long)&As[0][0],
                          (unsigned)(unsigned long long)&As[1][0] };
  stage_issue(As[0], lds_off[0], Ain  + (size_t)m0 * DIM, tid, wave);
  stage_issue(As[1], lds_off[1], Aout + (size_t)m0 * DIM, tid, wave);

  const float* bU[4];
  const float* bO[4];
  #pragma unroll
  for (int g = 0; g < 4; ++g) {
    bU[g] = WT + 4 * WT_MAT + (g * DIM + colL) * DIM + 2 * hi;
    bO[g] = WT + 5 * WT_MAT + (g * DIM + colL) * DIM + 2 * hi;
  }

  int rA0 = (mw * 32 + l15) * DIM + 2 * hi;
  int rA1 = rA0 + 16 * DIM;

  v8f acc[4][2] = {};

  stage_wait<1>(wave);
  __syncthreads();
  #pragma unroll 4
  for (int ks = 0; ks < DIM; ks += 4) {
    v2f a0 = *(const v2f*)(&As[0][0] + rA0 + ks);
    v2f a1 = *(const v2f*)(&As[0][0] + rA1 + ks);
    #pragma unroll
    for (int g = 0; g < 4; ++g) {
      v2f bv = *(const v2f*)(bU[g] + ks);
      acc[g][0] = wmma_f32(a0, bv, acc[g][0]);
      acc[g][1] = wmma_f32(a1, bv, acc[g][1]);
    }
  }

  stage_wait<0>(wave);
  __syncthreads();
  #pragma unroll 4
  for (int ks = 0; ks < DIM; ks += 4) {
    v2f a0 = *(const v2f*)(&As[1][0] + rA0 + ks);
    v2f a1 = *(const v2f*)(&As[1][0] + rA1 + ks);
    #pragma unroll
    for (int g = 0; g < 4; ++g) {
      v2f bv = *(const v2f*)(bO[g] + ks);
      acc[g][0] = wmma_f32(a0, bv, acc[g][0]);
      acc[g][1] = wmma_f32(a1, bv, acc[g][1]);
    }
  }

  #pragma unroll
  for (int sub = 0; sub < 2; ++sub) {
    #pragma unroll
    for (int i = 0; i < 8; ++i) {
      int r = m0 + mw * 32 + sub * 16 + i + 8 * hi;
      const float* zr = zc + r * ZCOLS + colL;
      float z0 = acc[0][sub][i] + zr[0];
      float z1 = acc[1][sub][i] + zr[256];
      float z2 = acc[2][sub][i] + zr[512];
      float z3 = acc[3][sub][i] + zr[768];
      float ig = 1.0f / (1.0f + expf(-z0));
      float fg = 1.0f / (1.0f + expf(-z1));
      float og = 1.0f / (1.0f + expf(-z2));
      float ci = tanhf(z3);
      float cold = first ? 0.0f : c_in[r * DIM + colL];
      float cn = fg * cold + ig * ci;
      float hn = og * tanhf(cn);
      float mm = maskv[r];
      hn *= mm; cn *= mm;
      h_out[r * DIM + colL] = hn;
      c_out[r * DIM + colL] = cn;
      if (h_out2) h_out2[r * DIM + colL] = hn;
    }
  }
}

// ---------------------------------------------------------------------------
// kernel_launch
// ---------------------------------------------------------------------------
extern "C" void kernel_launch(void* const* d_in, const int* in_sizes, int n_in,
                              void* d_out, int out_size, void* d_ws, size_t ws_size,
                              hipStream_t stream) {
  const float* node_reps = (const float*)d_in[0];
  const float* mask      = (const float*)d_in[1];
  const int*   in_idx    = (const int*)  d_in[2];
  const int*   in_edge   = (const int*)  d_in[3];
  const float* in_mask   = (const float*)d_in[4];
  const int*   out_idx   = (const int*)  d_in[5];
  const int*   out_edge  = (const int*)  d_in[6];
  const float* out_mask  = (const float*)d_in[7];
  const float* edge_emb  = (const float*)d_in[8];
  const float* Wi        = (const float*)d_in[9];
  const float* Wo        = (const float*)d_in[10];
  const float* Ui        = (const float*)d_in[11];
  const float* Uo        = (const float*)d_in[12];
  const float* bi        = (const float*)d_in[13];
  const float* bo        = (const float*)d_in[14];
  const float* bui       = (const float*)d_in[15];
  const float* buo       = (const float*)d_in[16];

  float* ws      = (float*)d_ws;
  float* A_in    = ws;
  float* A_out   = ws + 1 * (size_t)AGG_ELEM;
  float* E_in    = ws + 2 * (size_t)AGG_ELEM;
  float* E_out   = ws + 3 * (size_t)AGG_ELEM;
  float* z_const = ws + 4 * (size_t)AGG_ELEM;
  float* c_buf   = z_const + (size_t)M_ROWS * ZCOLS;
  float* WT      = c_buf + (size_t)AGG_ELEM;
  float* bias    = WT + 6 * (size_t)WT_MAT;

  float* outp  = (float*)d_out;
  float* reps0 = outp;
  float* reps1 = outp + 1 * (size_t)AGG_ELEM;
  float* out_h = outp + 2 * (size_t)AGG_ELEM;
  float* out_c = outp + 3 * (size_t)AGG_ELEM;

  prep_wt<<<(6 * WT_MAT) / 256, 256, 0, stream>>>(Wi, Wo, Ui, Uo, WT);
  prep_bias<<<ZCOLS / 256, 256, 0, stream>>>(bi, bo, bui, buo, bias);

  gather_agg<<<M_ROWS / 8, 256, 0, stream>>>(node_reps,
      in_idx, in_edge, in_mask, out_idx, out_edge, out_mask, edge_emb,
      A_in, A_out, E_in, E_out, 1);

  gemm_zconst<<<dim3(M_ROWS / TILE_M, ZCOLS / 64), 256, 0, stream>>>(
      A_in, E_in, A_out, E_out, WT, bias, z_const);

  gemm_lstm<<<dim3(M_ROWS / TILE_M, DIM / 64), 256, 0, stream>>>(
      A_in, A_out, WT, z_const, mask,
      nullptr, c_buf, reps0, nullptr, 1);

  gather_agg<<<M_ROWS / 8, 256, 0, stream>>>(reps0,
      in_idx, in_edge, in_mask, out_idx, out_edge, out_mask, edge_emb,
      A_in, A_out, nullptr, nullptr, 0);

  gemm_lstm<<<dim3(M_ROWS / TILE_M, DIM / 64), 256, 0, stream>>>(
      A_in, A_out, WT, z_const, mask,
      c_buf, out_c, reps1, out_h, 0);
}